// GaussianSplatting_13589276524714
// MI455X (gfx1250) — compile-verified
//
#include <hip/hip_runtime.h>

#define NGAUSS 2048
#define IMGSZ  128
#define FOCAL  128.0f
#define REC    12   // record: h0..h5, r, g, b, opv, 0, 0  (pads double as K=6,7 zeros)

typedef float v2f __attribute__((ext_vector_type(2)));
typedef float v8f __attribute__((ext_vector_type(8)));

__device__ __forceinline__ float sigmoidf(float x) { return 1.0f / (1.0f + __expf(-x)); }

// ---------------------------------------------------------------------------
// Kernel 1: per-Gaussian preprocessing -> record (blend params as rank-6
// features) + orderable depth sort key.
// ---------------------------------------------------------------------------
__global__ __launch_bounds__(256) void gs_preprocess(
    const float* __restrict__ means, const float* __restrict__ log_scales,
    const float* __restrict__ rotations, const float* __restrict__ opacity_logit,
    const float* __restrict__ color_pre, const float* __restrict__ cam_pos,
    const float* __restrict__ look_at,
    float* __restrict__ rec, unsigned long long* __restrict__ keys)
{
    int i = blockIdx.x * blockDim.x + threadIdx.x;
    if (i >= NGAUSS) return;

    // camera basis (redundant per thread, trivially cheap)
    float cx = cam_pos[0], cy = cam_pos[1], cz = cam_pos[2];
    float fx = look_at[0] - cx, fy = look_at[1] - cy, fz = look_at[2] - cz;
    float fl = rsqrtf(fx * fx + fy * fy + fz * fz);
    fx *= fl; fy *= fl; fz *= fl;
    // right = normalize(cross(fwd, [0,1,0]))
    float rx = -fz, ry = 0.0f, rz = fx;
    float rl = rsqrtf(rx * rx + ry * ry + rz * rz);
    rx *= rl; ry *= rl; rz *= rl;
    // up = cross(right, fwd)
    float ux = ry * fz - rz * fy;
    float uy = rz * fx - rx * fz;
    float uz = rx * fy - ry * fx;

    // quaternion -> rotation
    float qw = rotations[i * 4 + 0], qx = rotations[i * 4 + 1];
    float qy = rotations[i * 4 + 2], qz = rotations[i * 4 + 3];
    float qn = rsqrtf(qw * qw + qx * qx + qy * qy + qz * qz);
    qw *= qn; qx *= qn; qy *= qn; qz *= qn;
    float r00 = 1.f - 2.f * (qy * qy + qz * qz), r01 = 2.f * (qx * qy - qw * qz), r02 = 2.f * (qx * qz + qw * qy);
    float r10 = 2.f * (qx * qy + qw * qz), r11 = 1.f - 2.f * (qx * qx + qz * qz), r12 = 2.f * (qy * qz - qw * qx);
    float r20 = 2.f * (qx * qz - qw * qy), r21 = 2.f * (qy * qz + qw * qx), r22 = 1.f - 2.f * (qx * qx + qy * qy);

    float sx = __expf(log_scales[i * 3 + 0]);
    float sy = __expf(log_scales[i * 3 + 1]);
    float sz = __expf(log_scales[i * 3 + 2]);

    // M = R * diag(s);  cov3d = M M^T (symmetric)
    float m00 = r00 * sx, m01 = r01 * sy, m02 = r02 * sz;
    float m10 = r10 * sx, m11 = r11 * sy, m12 = r12 * sz;
    float m20 = r20 * sx, m21 = r21 * sy, m22 = r22 * sz;
    float c00 = m00 * m00 + m01 * m01 + m02 * m02;
    float c01 = m00 * m10 + m01 * m11 + m02 * m12;
    float c02 = m00 * m20 + m01 * m21 + m02 * m22;
    float c11 = m10 * m10 + m11 * m11 + m12 * m12;
    float c12 = m10 * m20 + m11 * m21 + m12 * m22;
    float c22 = m20 * m20 + m21 * m21 + m22 * m22;

    // camera-space mean / depth / 2D mean
    float dxm = means[i * 3 + 0] - cx, dym = means[i * 3 + 1] - cy, dzm = means[i * 3 + 2] - cz;
    float mcx = rx * dxm + ry * dym + rz * dzm;
    float mcy = ux * dxm + uy * dym + uz * dzm;
    float depth = fx * dxm + fy * dym + fz * dzm;   // = -means_cam.z
    float invz = 1.0f / (depth + 1e-8f);
    float m2dx = (mcx * invz) * FOCAL + (float)IMGSZ * 0.5f;
    float m2dy = -(mcy * invz) * FOCAL + (float)IMGSZ * 0.5f;

    // 2x2 of Wm C Wm^T  (rows right, up)
    float v0x = c00 * rx + c01 * ry + c02 * rz;
    float v0y = c01 * rx + c11 * ry + c12 * rz;
    float v0z = c02 * rx + c12 * ry + c22 * rz;
    float a = rx * v0x + ry * v0y + rz * v0z;
    float b = ux * v0x + uy * v0y + uz * v0z;
    float v1x = c00 * ux + c01 * uy + c02 * uz;
    float v1y = c01 * ux + c11 * uy + c12 * uz;
    float v1z = c02 * ux + c12 * uy + c22 * uz;
    float d = ux * v1x + uy * v1y + uz * v1z;

    float zsq = fmaxf(depth * depth, 0.01f);
    float s2 = (FOCAL * FOCAL) / zsq;
    float a2 = a * s2 + 0.3f, b2 = b * s2, d2 = d * s2 + 0.3f;
    float det = fmaxf(a2 * d2 - b2 * b2, 1e-6f);
    float rdet = 1.0f / det;
    float ia = d2 * rdet, ib = -b2 * rdet, idd = a2 * rdet;

    int valid = (depth > 0.1f) && (m2dx > -(float)IMGSZ) && (m2dx < 2.f * IMGSZ) &&
                (m2dy > -(float)IMGSZ) && (m2dy < 2.f * IMGSZ);
    float opv = sigmoidf(opacity_logit[i]) * (valid ? 1.0f : 0.0f);

    // rank-6 Gaussian features: mahal = f(p) . h(g)
    float* R = rec + (size_t)i * REC;
    R[0] = ia;
    R[1] = -2.0f * (ia * m2dx + ib * m2dy);
    R[2] = idd;
    R[3] = -2.0f * (idd * m2dy + ib * m2dx);
    R[4] = 2.0f * ib;
    R[5] = ia * m2dx * m2dx + 2.0f * ib * m2dx * m2dy + idd * m2dy * m2dy;
    R[6] = sigmoidf(color_pre[i * 3 + 0]);
    R[7] = sigmoidf(color_pre[i * 3 + 1]);
    R[8] = sigmoidf(color_pre[i * 3 + 2]);
    R[9] = opv;
    R[10] = 0.0f;   // K=6 zero row for high-half B operand
    R[11] = 0.0f;   // K=7 zero row

    // order-preserving key (stable: ties broken by index, matching jnp.argsort)
    unsigned u = __float_as_uint(depth);
    u = (u & 0x80000000u) ? ~u : (u | 0x80000000u);
    keys[i] = (((unsigned long long)u) << 32) | (unsigned)i;
}

// ---------------------------------------------------------------------------
// Kernel 2: single-block bitonic sort of 2048 (depth|idx) keys in LDS.
// ---------------------------------------------------------------------------
__global__ __launch_bounds__(1024) void gs_sort(const unsigned long long* __restrict__ keys,
                                                unsigned* __restrict__ sidx)
{
    __shared__ unsigned long long k[NGAUSS];
    int t = threadIdx.x;
    for (int i = t; i < NGAUSS; i += 1024) k[i] = keys[i];
    __syncthreads();
    for (int kk = 2; kk <= NGAUSS; kk <<= 1) {
        for (int j = kk >> 1; j > 0; j >>= 1) {
            for (int i = t; i < NGAUSS; i += 1024) {
                int ixj = i ^ j;
                if (ixj > i) {
                    bool up = ((i & kk) == 0);
                    unsigned long long x = k[i], y = k[ixj];
                    if ((x > y) == up) { k[i] = y; k[ixj] = x; }
                }
            }
            __syncthreads();
        }
    }
    for (int i = t; i < NGAUSS; i += 1024) sidx[i] = (unsigned)(k[i] & 0xFFFFFFFFull);
}

// ---------------------------------------------------------------------------
// Kernel 3: tiled rasterizer. 16x16 pixel tile per block (1 lane = 1 pixel).
// Per 16-Gaussian batch: stage records to LDS; each wave computes its
// 2x (16px x 16G) Mahalanobis tiles with chained V_WMMA_F32_16X16X4_F32
// (K padded 6->8), transposes D through LDS, then VALU does exp + blend.
// A operands and all LDS addresses are loop-invariant (hoisted); B operands
// are branchless ds_load_2addr from fixed per-lane columns (zero pads give
// the K=6,7 rows for the high half).
// ---------------------------------------------------------------------------
__global__ __launch_bounds__(256) void gs_raster(const float* __restrict__ rec,
                                                 const unsigned* __restrict__ sidx,
                                                 float* __restrict__ out)
{
    __shared__ float hfeat[16][REC];      // batch of 16 gaussians
    __shared__ float mtile[8][32][17];    // per-wave mahal tile, padded pitch

    int t = threadIdx.x;
    int w = t >> 5, L = t & 31;
    int bx = (blockIdx.x & 7) * 16;
    int by = (blockIdx.x >> 3) * 16;
    int lx = L & 15;
    int half = L >> 4;                    // 0 = lanes 0-15, 1 = lanes 16-31
    int px = bx + lx, py = by + 2 * w + half;

    float imr = 1.0f, img = 1.0f, imb = 1.0f, T = 1.0f;

    // ---- loop-invariant A operands: f = [x^2, x, y^2, y, xy, 1, 0, 0] ----
    float xm  = (float)(bx + lx);         // A-matrix column M = L%16
    float sel = half ? 0.0f : 1.0f;
    v2f A0[2], A1[2];
#pragma unroll
    for (int rt = 0; rt < 2; ++rt) {
        float yy = (float)(by + 2 * w + rt);
        float u  = half ? yy : xm;        // K rows {0,1} (low) / {2,3} (high)
        A0[rt].x = u * u;
        A0[rt].y = u;
        A1[rt].x = xm * yy * sel;         // K=4 (xy) low half, K=6 (0) high
        A1[rt].y = sel;                   // K=5 (1)  low half, K=7 (0) high
    }
    // loop-invariant per-lane B columns in hfeat
    int gb = L & 15;
    int i0 = half * 2;                    // K rows {0,1} | {2,3}
    int i1 = 4 + half * 6;                // K rows {4,5} | {10,11}==0
    // loop-invariant D-tile store rows
    int mb0 = half * 8;                   // tile rt=0
    int mb1 = 16 + half * 8;              // tile rt=1

    for (int base = 0; base < NGAUSS; base += 16) {
        __syncthreads();   // previous batch done reading hfeat/mtile
        if (t < 16 * REC) {
            int g = t / REC, f = t - g * REC;
            unsigned gid = sidx[base + g];
            hfeat[g][f] = rec[(size_t)gid * REC + f];
        }
        __syncthreads();

        v2f b0, b1;
        b0.x = hfeat[gb][i0];
        b0.y = hfeat[gb][i0 + 1];
        b1.x = hfeat[gb][i1];
        b1.y = hfeat[gb][i1 + 1];

        v8f d0 = {};
        d0 = __builtin_amdgcn_wmma_f32_16x16x4_f32(false, A0[0], false, b0,
                                                   (short)0, d0, false, false);
        d0 = __builtin_amdgcn_wmma_f32_16x16x4_f32(false, A1[0], false, b1,
                                                   (short)0, d0, false, false);
        v8f d1 = {};
        d1 = __builtin_amdgcn_wmma_f32_16x16x4_f32(false, A0[1], false, b0,
                                                   (short)0, d1, false, false);
        d1 = __builtin_amdgcn_wmma_f32_16x16x4_f32(false, A1[1], false, b1,
                                                   (short)0, d1, false, false);
#pragma unroll
        for (int c = 0; c < 8; ++c) {
            mtile[w][mb0 + c][gb] = d0[c];
            mtile[w][mb1 + c][gb] = d1[c];
        }
        __syncthreads();

        // blend: lane's within-wave pixel index == L (conflict-free: pitch 17)
#pragma unroll 4
        for (int g = 0; g < 16; ++g) {
            float mah = mtile[w][L][g];
            float al = fminf(hfeat[g][9] * __expf(-0.5f * mah), 0.99f);
            float Ta = T * al;
            imr += Ta * (hfeat[g][6] - imr);
            img += Ta * (hfeat[g][7] - img);
            imb += Ta * (hfeat[g][8] - imb);
            T *= (1.0f - al);
        }
    }

    size_t o = ((size_t)py * IMGSZ + px) * 3;
    out[o + 0] = fminf(fmaxf(imr, 0.0f), 1.0f);
    out[o + 1] = fminf(fmaxf(img, 0.0f), 1.0f);
    out[o + 2] = fminf(fmaxf(imb, 0.0f), 1.0f);
}

extern "C" void kernel_launch(void* const* d_in, const int* in_sizes, int n_in,
                              void* d_out, int out_size, void* d_ws, size_t ws_size,
                              hipStream_t stream) {
    const float* means         = (const float*)d_in[0];
    const float* log_scales    = (const float*)d_in[1];
    const float* rotations     = (const float*)d_in[2];
    const float* opacity_logit = (const float*)d_in[3];
    const float* color_pre     = (const float*)d_in[4];
    const float* cam_pos       = (const float*)d_in[5];
    const float* look_at       = (const float*)d_in[6];
    float* out = (float*)d_out;

    float* rec = (float*)d_ws;                                             // 2048*12 f32
    unsigned long long* keys =
        (unsigned long long*)((char*)d_ws + (size_t)NGAUSS * REC * sizeof(float));
    unsigned* sidx = (unsigned*)((char*)keys + (size_t)NGAUSS * sizeof(unsigned long long));

    gs_preprocess<<<NGAUSS / 256, 256, 0, stream>>>(means, log_scales, rotations,
                                                    opacity_logit, color_pre,
                                                    cam_pos, look_at, rec, keys);
    gs_sort<<<1, 1024, 0, stream>>>(keys, sidx);
    gs_raster<<<(IMGSZ / 16) * (IMGSZ / 16), 256, 0, stream>>>(rec, sidx, out);
}